// MCPBRNN_Generic_constant_Out_variableLoss_BYPASSM0_53695681134736
// MI455X (gfx1250) — compile-verified
//
#include <hip/hip_runtime.h>
#include <math.h>

typedef __attribute__((ext_vector_type(2))) float v2f;
typedef __attribute__((ext_vector_type(8))) float v8f;

#define SPIN_N   1000
#define TRAIN_N  800000
#define ML_C     2.9086f
#define SL_C     1.898f
#define CHUNK    128   // outputs per thread in the scan
#define BURN     128   // warm-up steps; Lipschitz <= 0.845 -> 0.845^128 ~ 4e-10

// ---------------------------------------------------------------------------
// Wave-wide f32 sum using V_WMMA_F32_16X16X4_F32.
// A (16x4) carries the 32 lane values in VGPR0 (lane m -> A[m][0], lane m+16
// -> A[m][2]), VGPR1 = 0.  B = ones(4x16).  D[m][n] = v[m] + v[m+16].
// Summing the 8 D components on a lane gives the half-wave total; one
// shfl_xor(16) completes the full 32-lane sum on every lane.
// Must execute with EXEC all ones (convergent control flow).
// ---------------------------------------------------------------------------
__device__ __forceinline__ float wave_sum32_wmma(float v) {
    v2f a; a[0] = v;    a[1] = 0.0f;
    v2f b; b[0] = 1.0f; b[1] = 1.0f;
    v8f c = {0.f, 0.f, 0.f, 0.f, 0.f, 0.f, 0.f, 0.f};
    v8f d = __builtin_amdgcn_wmma_f32_16x16x4_f32(
        /*neg_a=*/false, a, /*neg_b=*/false, b,
        /*c_mod=*/(short)0, c, /*reuse_a=*/false, /*reuse_b=*/false);
    float t = d[0] + d[1] + d[2] + d[3] + d[4] + d[5] + d[6] + d[7];
    t += __shfl_xor(t, 16, 32);
    return t;
}

__global__ void init_ws_kernel(double* sums) {
    sums[0] = 0.0;
    sums[1] = 0.0;
}

// Grid-stride sum / sum-of-squares over y_obs[SPIN_N:TRAIN_N), wave-reduced
// via WMMA, accumulated in f64 atomics.
__global__ void obsstd_reduce_kernel(const float* __restrict__ y,
                                     double* __restrict__ sums, int ylen) {
    int idx    = blockIdx.x * blockDim.x + threadIdx.x;
    int stride = gridDim.x * blockDim.x;
    int hi     = (TRAIN_N < ylen) ? TRAIN_N : ylen;
    float s = 0.0f, q = 0.0f;
    for (int i = SPIN_N + idx; i < hi; i += stride) {
        float v = y[i];
        s += v;
        q = fmaf(v, v, q);
    }
    float ts = wave_sum32_wmma(s);
    float tq = wave_sum32_wmma(q);
    if ((threadIdx.x & 31) == 0) {
        atomicAdd(&sums[0], (double)ts);
        atomicAdd(&sums[1], (double)tq);
    }
}

__global__ void obsstd_finalize_kernel(const double* __restrict__ sums,
                                       float* __restrict__ std_out, int ylen) {
    int hi = (TRAIN_N < ylen) ? TRAIN_N : ylen;
    double n    = (double)(hi - SPIN_N);
    double mean = sums[0] / n;
    double var  = (sums[1] - n * mean * mean) / (n - 1.0);  // ddof=1
    std_out[0] = (float)sqrt(var > 0.0 ? var : 0.0);
}

// ---------------------------------------------------------------------------
// Chunk-parallel burn-in scan of the contractive recurrence.
// Each thread: warm-start BURN steps from c=0, then produce CHUNK outputs.
// Out layout (floats): [0,N) h | [N,2N) c | [2N,3N) l | [3N,4N) bp |
// [4N,5N) ib | [5N,6N) oo | [6N,7N) ol | [7N,8N) f |
// [8N,10N) h_nout interleaved (h, obsstd) | [10N,11N) obs_std.
// ---------------------------------------------------------------------------
__global__ void mcpbrnn_scan_kernel(
    const float2* __restrict__ x,   // (N, 2) as float2 per step
    const float* __restrict__ w_yom, const float* __restrict__ w_ylm,
    const float* __restrict__ w_yfm, const float* __restrict__ bias_p,
    const float* __restrict__ wb2_p, const float* __restrict__ theltaC_p,
    const int*   __restrict__ tl_p,  const float* __restrict__ std_p,
    float* __restrict__ out, int N)
{
    long long tid   = (long long)blockIdx.x * blockDim.x + threadIdx.x;
    long long start = tid * CHUNK;
    if (start >= (long long)N) return;
    long long end = start + CHUNK;
    if (end > (long long)N) end = N;

    // Scalar gate parameters (H == 1).
    float eo   = __expf(w_yom[0]);
    float el   = __expf(w_ylm[0]);
    float ef   = __expf(w_yfm[0]);
    float den  = eo + el + ef;
    float oo   = eo / den;
    float ol1  = el / den;
    float expT = __expf(theltaC_p[0]);
    float bias = bias_p[0];
    float wb2  = wb2_p[0];
    int   tl   = tl_p[0];
    float obsstd = std_p[0];

    float c = 0.0f;

    // Warm-up: converge onto the true trajectory (exact when bstart == tl).
    long long bstart = start - BURN;
    if (bstart < (long long)tl) bstart = (long long)tl;
    for (long long i = bstart; i < start; ++i) {
        float2 u = x[i];
        float ol = ol1 / (1.0f + __expf(-(bias + ((u.y - ML_C) / SL_C) * wb2)));
        float f  = 1.0f - oo - ol;
        float px = fmaxf(u.x + c - expT, 0.0f);
        float ib = (u.x == 0.0f) ? 0.0f : (px / u.x);
        c = f * c + (1.0f - ib) * u.x;
    }

    const long long Nl = (long long)N;
    for (long long i = start; i < end; ++i) {
        if (i < (long long)tl) {
            out[i]              = 0.0f;
            out[Nl + i]         = 0.0f;
            out[2 * Nl + i]     = 0.0f;
            out[3 * Nl + i]     = 0.0f;
            out[4 * Nl + i]     = 0.0f;
            out[5 * Nl + i]     = 0.0f;
            out[6 * Nl + i]     = 0.0f;
            out[7 * Nl + i]     = 0.0f;
            out[8 * Nl + 2 * i]     = 0.0f;
            out[8 * Nl + 2 * i + 1] = 0.0f;
            out[10 * Nl + i]    = 0.0f;
            continue;  // carry unchanged for invalid steps
        }
        float2 u = x[i];
        float ol = ol1 / (1.0f + __expf(-(bias + ((u.y - ML_C) / SL_C) * wb2)));
        float f  = 1.0f - oo - ol;
        float px = fmaxf(u.x + c - expT, 0.0f);
        float ib = (u.x == 0.0f) ? 0.0f : (px / u.x);
        float bp = ib * u.x;
        float h  = oo * c + bp;
        float l  = ol * c;

        out[i]              = h;
        out[Nl + i]         = c;      // pre-update carry
        out[2 * Nl + i]     = l;
        out[3 * Nl + i]     = bp;
        out[4 * Nl + i]     = ib;
        out[5 * Nl + i]     = oo;
        out[6 * Nl + i]     = ol;
        out[7 * Nl + i]     = f;
        out[8 * Nl + 2 * i]     = h;       // h_nout[:,0]
        out[8 * Nl + 2 * i + 1] = obsstd;  // h_nout[:,1]
        out[10 * Nl + i]    = obsstd;      // obs_std

        c = f * c + (1.0f - ib) * u.x;
    }
}

extern "C" void kernel_launch(void* const* d_in, const int* in_sizes, int n_in,
                              void* d_out, int out_size, void* d_ws, size_t ws_size,
                              hipStream_t stream) {
    (void)n_in; (void)out_size; (void)ws_size;

    const float2* x       = (const float2*)d_in[0];
    const float* y_obs    = (const float*)d_in[1];
    const float* w_yom    = (const float*)d_in[2];
    const float* w_ylm    = (const float*)d_in[3];
    const float* w_yfm    = (const float*)d_in[4];
    const float* bias     = (const float*)d_in[5];
    const float* wb2      = (const float*)d_in[6];
    const float* theltaC  = (const float*)d_in[7];
    /* d_in[8] = epoch, unused by the forward */
    const int*   time_lag = (const int*)d_in[9];
    float* out = (float*)d_out;

    int N    = in_sizes[0] / 2;  // x is (B*T, 2)
    int ylen = in_sizes[1];

    double* sums = (double*)d_ws;
    float*  stdp = (float*)((char*)d_ws + 2 * sizeof(double));

    hipLaunchKernelGGL(init_ws_kernel, dim3(1), dim3(1), 0, stream, sums);
    hipLaunchKernelGGL(obsstd_reduce_kernel, dim3(128), dim3(256), 0, stream,
                       y_obs, sums, ylen);
    hipLaunchKernelGGL(obsstd_finalize_kernel, dim3(1), dim3(1), 0, stream,
                       sums, stdp, ylen);

    long long nchunks = ((long long)N + CHUNK - 1) / CHUNK;
    int blocks = (int)((nchunks + 255) / 256);
    hipLaunchKernelGGL(mcpbrnn_scan_kernel, dim3(blocks), dim3(256), 0, stream,
                       x, w_yom, w_ylm, w_yfm, bias, wb2, theltaC,
                       time_lag, stdp, out, N);
}